// Embedding_69604239999332
// MI455X (gfx1250) — compile-verified
//
#include <hip/hip_runtime.h>
#include <math.h>

// CDNA5 / gfx1250: wave32, WMMA. One wave computes a 16x16 f32 tile with
// V_WMMA_F32_16X16X4_F32 (f32 A/B frags = 2 VGPRs each, C/D = 8 VGPRs).
typedef __attribute__((ext_vector_type(2))) float v2f;
typedef __attribute__((ext_vector_type(8))) float v8f;

#define TN 8
#define TE 8
#define SROW 1028  // padded LDS row stride in floats (1024 + 4): rows hit distinct banks

__device__ __forceinline__ int imin(int a, int b) { return a < b ? a : b; }

__device__ __forceinline__ void f4add(float4& a, const float4& v) {
  a.x += v.x; a.y += v.y; a.z += v.z; a.w += v.w;
}

__global__ void zero_i32_kernel(int* __restrict__ p, int n) {
  int i = blockIdx.x * blockDim.x + threadIdx.x;
  if (i < n) p[i] = 0;
}

__global__ void hist_kernel(const int* __restrict__ key, int* __restrict__ counts, int n) {
  int i = blockIdx.x * blockDim.x + threadIdx.x;
  if (i < n) atomicAdd(&counts[key[i]], 1);
}

// Exclusive scan of cnt[0..N) -> rowp[0..N], plus a second copy in cursor.
// One block, 64 threads, chunked sequential scan (runs once per call).
__global__ void scan_kernel(const int* __restrict__ cnt, int* __restrict__ rowp,
                            int* __restrict__ cursor, int N) {
  __shared__ int csum[64];
  const int chunk = (N + 63) / 64;
  const int t = threadIdx.x;
  const int b = t * chunk, e = imin(b + chunk, N);
  int s = 0;
  for (int i = b; i < e; ++i) s += cnt[i];
  csum[t] = s;
  __syncthreads();
  if (t == 0) {
    int acc = 0;
    for (int i = 0; i < 64; ++i) { int v = csum[i]; csum[i] = acc; acc += v; }
  }
  __syncthreads();
  int acc = csum[t];
  for (int i = b; i < e; ++i) { int v = cnt[i]; rowp[i] = acc; cursor[i] = acc; acc += v; }
  if (t == 63) rowp[N] = acc;
}

// Counting-sort edges by dst: emit src and etype in dst-sorted order.
__global__ void edge_sort_kernel(const int* __restrict__ src, const int* __restrict__ dst,
                                 const int* __restrict__ et, int* __restrict__ cursor,
                                 int* __restrict__ sSrc, int* __restrict__ sT, int E) {
  int e = blockIdx.x * blockDim.x + threadIdx.x;
  if (e < E) {
    int pos = atomicAdd(&cursor[dst[e]], 1);
    sSrc[pos] = src[e];
    sT[pos] = et[e];
  }
}

// Single-thread: prefix-sum node-type counts, emit per-16-row tile descriptors
// for the segmented (typed) GEMMs. <= ceil(N/16)+TN tiles.
__global__ void build_tiles_kernel(const int* __restrict__ counts, int* __restrict__ cursor,
                                   int* __restrict__ tType, int* __restrict__ tStart,
                                   int* __restrict__ tCnt, int* __restrict__ numTiles) {
  if (threadIdx.x != 0 || blockIdx.x != 0) return;
  int off = 0, nt = 0;
  for (int t = 0; t < TN; ++t) {
    cursor[t] = off;
    int c = counts[t];
    for (int s = 0; s < c; s += 16) {
      tType[nt] = t; tStart[nt] = off + s; tCnt[nt] = imin(16, c - s); ++nt;
    }
    off += c;
  }
  *numTiles = nt;
}

__global__ void scatter_perm_kernel(const int* __restrict__ types, int* __restrict__ cursor,
                                    int* __restrict__ perm, int n) {
  int i = blockIdx.x * blockDim.x + threadIdx.x;
  if (i < n) {
    int pos = atomicAdd(&cursor[types[i]], 1);
    perm[pos] = i;
  }
}

// Fused RelGraphConv for 16 dst rows per block (H == 128, blockDim == 256):
//   msg[d,:] = tanh( S[d,:] @ Wrel + h[d,:] @ Wloop + bias )
// where S[d, t*H+c] = sum over type-t in-edges of h[src, c].
// Phase 1: each wave aggregates 2 dst rows (8 float4 per-type register
// accumulators, wave-uniform switch on etype) -> 16x1024 LDS tile, no atomics.
// Phase 2: 8 waves x one 16x16 column tile, K=1024 WMMA with A from LDS
// (ds_load_b64) + fused K=128 self-loop GEMM + bias + tanh epilogue.
__global__ void relconv_kernel(const float* __restrict__ h,
                               const int* __restrict__ rowp,
                               const int* __restrict__ sSrc,
                               const int* __restrict__ sT,
                               const float* __restrict__ Wrel,   // (TE*H, H) row-major
                               const float* __restrict__ Wloop,  // (H, H)
                               const float* __restrict__ bias,   // (H)
                               float* __restrict__ msg, int N) {
  extern __shared__ float Sl[];  // 16 * SROW floats
  const int rowBase = blockIdx.x * 16;
  const int wave = threadIdx.x >> 5;
  const int lane = threadIdx.x & 31;

  // ---- phase 1: typed aggregation ----
  for (int r = wave * 2; r < wave * 2 + 2; ++r) {
    const int d = rowBase + r;
    float4 a0 = {0,0,0,0}, a1 = a0, a2 = a0, a3 = a0, a4 = a0, a5 = a0, a6 = a0, a7 = a0;
    if (d < N) {
      const int i1 = rowp[d + 1];
      for (int i = rowp[d]; i < i1; ++i) {
        const int s = sSrc[i];
        const int t = sT[i];  // uniform across the wave
        const float4 v = *(const float4*)(h + (long)s * 128 + lane * 4);
        switch (t) {
          case 0: f4add(a0, v); break;
          case 1: f4add(a1, v); break;
          case 2: f4add(a2, v); break;
          case 3: f4add(a3, v); break;
          case 4: f4add(a4, v); break;
          case 5: f4add(a5, v); break;
          case 6: f4add(a6, v); break;
          default: f4add(a7, v); break;
        }
      }
    }
    float* row = Sl + r * SROW + lane * 4;
    *(float4*)(row + 0 * 128) = a0; *(float4*)(row + 1 * 128) = a1;
    *(float4*)(row + 2 * 128) = a2; *(float4*)(row + 3 * 128) = a3;
    *(float4*)(row + 4 * 128) = a4; *(float4*)(row + 5 * 128) = a5;
    *(float4*)(row + 6 * 128) = a6; *(float4*)(row + 7 * 128) = a7;
  }
  __syncthreads();

  // ---- phase 2: WMMA GEMM ----
  const int mi = lane & 15;
  const int half = lane >> 4;
  const int n = wave * 16 + mi;  // Odim = 128, 8 waves cover all columns
  v8f acc = {};

  const float* al = Sl + mi * SROW + 2 * half;        // A frag from LDS
  const float* w1 = Wrel + (long)(2 * half) * 128 + n;
#pragma unroll 4
  for (int k0 = 0; k0 < TE * 128; k0 += 4) {
    v2f a = *(const v2f*)(al + k0);
    v2f b;
    b.x = w1[(long)k0 * 128];
    b.y = w1[(long)(k0 + 1) * 128];
    acc = __builtin_amdgcn_wmma_f32_16x16x4_f32(false, a, false, b, (short)0, acc,
                                                false, false);
  }
  // fused self-loop: + h @ Wloop
  const int dA = imin(rowBase + mi, N - 1);
  const float* a2p = h + (long)dA * 128 + 2 * half;
  const float* w2 = Wloop + (long)(2 * half) * 128 + n;
#pragma unroll 4
  for (int k0 = 0; k0 < 128; k0 += 4) {
    v2f a = *(const v2f*)(a2p + k0);
    v2f b;
    b.x = w2[(long)k0 * 128];
    b.y = w2[(long)(k0 + 1) * 128];
    acc = __builtin_amdgcn_wmma_f32_16x16x4_f32(false, a, false, b, (short)0, acc,
                                                false, false);
  }

  const float bi = bias[n];
#pragma unroll
  for (int v = 0; v < 8; ++v) {
    const int d = rowBase + v + 8 * half;
    if (d < N) msg[(long)d * 128 + n] = tanhf(acc[v] + bi);
  }
}

// Y[node, :] = act( X1 @ W1[type] + X2 @ W2[type] + bias )
// W1/W2 row-major (k1 x Odim) / (k2 x Odim); wStride = per-type element stride
// (applied to both; 0 => untyped). perm/tType/tStart/tCnt/numTiles give the
// type-segmented row-tile mapping (nullptr => identity rows). blockDim = 256;
// wave w -> column tile blockIdx.y*8+w so all 8 waves reuse the same A rows
// through the WGP cache. All early exits are wave-uniform: EXEC is all-ones at
// every WMMA (ISA 7.12 requirement).
__global__ void wmma_gemm_kernel(
    const float* __restrict__ X1, int k1,
    const float* __restrict__ X2, int k2,
    const float* __restrict__ W1b, const float* __restrict__ W2b, long wStride,
    const float* __restrict__ bias,
    float* __restrict__ Y, int Odim, int Nrows,
    const int* __restrict__ perm, const int* __restrict__ tType,
    const int* __restrict__ tStart, const int* __restrict__ tCnt,
    const int* __restrict__ numTiles, int act) {
  const int rt = blockIdx.x;
  if (numTiles && rt >= *numTiles) return;
  const int wave = threadIdx.x >> 5;
  const int ct = blockIdx.y * 8 + wave;
  if (ct * 16 >= Odim) return;
  const int lane = threadIdx.x & 31;
  const int mi = lane & 15;
  const int half = lane >> 4;

  const int rowStart = tStart ? tStart[rt] : rt * 16;
  const int rowCnt = tCnt ? tCnt[rt] : imin(16, Nrows - rowStart);
  const int rA = rowStart + (mi < rowCnt ? mi : 0);
  const int nodeA = perm ? perm[rA] : rA;
  const long tOff = tType ? (long)tType[rt] * wStride : 0;
  const int n = ct * 16 + mi;

  v8f acc = {};

  const float* a1 = X1 + (long)nodeA * k1 + 2 * half;
  const float* w1 = W1b + tOff + (long)(2 * half) * Odim + n;
#pragma unroll 4
  for (int k0 = 0; k0 < k1; k0 += 4) {
    v2f a = *(const v2f*)(a1 + k0);
    v2f b;
    b.x = w1[(long)k0 * Odim];
    b.y = w1[(long)(k0 + 1) * Odim];
    acc = __builtin_amdgcn_wmma_f32_16x16x4_f32(false, a, false, b, (short)0, acc,
                                                false, false);
  }
  if (X2) {
    const float* a2 = X2 + (long)nodeA * k2 + 2 * half;
    const float* w2 = W2b + tOff + (long)(2 * half) * Odim + n;
#pragma unroll 4
    for (int k0 = 0; k0 < k2; k0 += 4) {
      v2f a = *(const v2f*)(a2 + k0);
      v2f b;
      b.x = w2[(long)k0 * Odim];
      b.y = w2[(long)(k0 + 1) * Odim];
      acc = __builtin_amdgcn_wmma_f32_16x16x4_f32(false, a, false, b, (short)0, acc,
                                                  false, false);
    }
  }

  const float bi = bias ? bias[n] : 0.0f;
#pragma unroll
  for (int v = 0; v < 8; ++v) {
    const int ml = v + 8 * half;
    if (ml < rowCnt) {
      const int r = rowStart + ml;
      const int node = perm ? perm[r] : r;
      float val = acc[v] + bi;
      if (act) val = tanhf(val);
      Y[(long)node * Odim + n] = val;
    }
  }
}

extern "C" void kernel_launch(void* const* d_in, const int* in_sizes, int n_in,
                              void* d_out, int out_size, void* d_ws, size_t ws_size,
                              hipStream_t stream) {
  const float* x = (const float*)d_in[0];
  const int* src = (const int*)d_in[1];
  const int* dst = (const int*)d_in[2];
  const int* etype = (const int*)d_in[3];
  const int* ntype = (const int*)d_in[4];
  const float* W_in = (const float*)d_in[5];
  const float* b_in = (const float*)d_in[6];
  const float* W_rel = (const float*)d_in[7];
  const float* W_loop = (const float*)d_in[8];
  const float* b_rel = (const float*)d_in[9];
  const float* W_up1 = (const float*)d_in[10];
  const float* W_up2 = (const float*)d_in[11];

  const int N = in_sizes[4];
  const int E = in_sizes[1];
  const int F = in_sizes[0] / N;  // 32
  const int H = in_sizes[6];      // 128
  const int L = in_sizes[9] / H;  // 12
  const long HH = (long)H * H;
  const int maxTiles = (N + 15) / 16 + TN;
  const int nTiles = (N + 15) / 16;

  // Workspace layout; everything read is written first each call.
  float* hA = (float*)d_ws;                     // N*H  features (ping)
  float* hB = hA + (size_t)N * H;               // N*H  features (pong)
  float* msg = hB + (size_t)N * H;              // N*H  RelGraphConv output
  float* mid = msg + (size_t)N * H;             // N*2H TypedLinear1 output
  int* rowp = (int*)(mid + (size_t)N * 2 * H);  // N+1  CSR row pointers (by dst)
  int* ecur = rowp + (N + 1);                   // N    edge-sort cursors / counts
  int* sSrc = ecur + N;                         // E    src, dst-sorted
  int* sT = sSrc + E;                           // E    etype, dst-sorted
  int* perm = sT + E;                           // N    type-sorted node order
  int* counts = perm + N;                       // TN
  int* curT = counts + TN;                      // TN
  int* numTiles = curT + TN;                    // 1
  int* tType = numTiles + 1;                    // maxTiles
  int* tStart = tType + maxTiles;               // maxTiles
  int* tCnt = tStart + maxTiles;                // maxTiles
  (void)ws_size; (void)n_in; (void)out_size;

  // ---- edge CSR by dst (graph is layer-invariant: once per call) ----
  zero_i32_kernel<<<(N + 255) / 256, 256, 0, stream>>>(ecur, N);
  hist_kernel<<<(E + 255) / 256, 256, 0, stream>>>(dst, ecur, E);
  scan_kernel<<<1, 64, 0, stream>>>(ecur, rowp, ecur, N);
  edge_sort_kernel<<<(E + 255) / 256, 256, 0, stream>>>(src, dst, etype, ecur, sSrc, sT, E);

  // ---- node-type grouping (once per call) ----
  zero_i32_kernel<<<1, 32, 0, stream>>>(counts, TN);
  hist_kernel<<<(N + 255) / 256, 256, 0, stream>>>(ntype, counts, N);
  build_tiles_kernel<<<1, 1, 0, stream>>>(counts, curT, tType, tStart, tCnt, numTiles);
  scatter_perm_kernel<<<(N + 255) / 256, 256, 0, stream>>>(ntype, curT, perm, N);

  // ---- input projection: hA = tanh(x @ W_in + b_in) ----
  wmma_gemm_kernel<<<dim3(nTiles, 1), 256, 0, stream>>>(
      x, F, nullptr, 0, W_in, nullptr, 0, b_in, hA, H, N,
      nullptr, nullptr, nullptr, nullptr, nullptr, 1);

  const size_t ldsBytes = (size_t)16 * SROW * sizeof(float);  // 65792 B
  float* h = hA;
  float* hn = hB;
  for (int l = 0; l < L; ++l) {
    // msg = tanh( S @ W_rel[l] + h @ W_loop[l] + b_rel[l] ), fused, no atomics.
    relconv_kernel<<<nTiles, 256, ldsBytes, stream>>>(
        h, rowp, sSrc, sT, W_rel + (long)l * TE * HH, W_loop + (long)l * HH,
        b_rel + (long)l * H, msg, N);

    // mid = tanh(TypedLinear1([h, msg]))  (K=2H, O=2H, segmented by node type)
    wmma_gemm_kernel<<<dim3(maxTiles, 2), 256, 0, stream>>>(
        h, H, msg, H,
        W_up1 + (long)l * TN * 4 * HH, W_up1 + (long)l * TN * 4 * HH + (long)H * 2 * H,
        4 * HH, nullptr, mid, 2 * H, N,
        perm, tType, tStart, tCnt, numTiles, 1);

    // h_next = tanh(TypedLinear2([h, mid]))  (K=3H, O=H); last layer -> d_out.
    float* Y = (l == L - 1) ? (float*)d_out : hn;
    wmma_gemm_kernel<<<dim3(maxTiles, 1), 256, 0, stream>>>(
        h, H, mid, 2 * H,
        W_up2 + (long)l * TN * 3 * HH, W_up2 + (long)l * TN * 3 * HH + HH,
        3 * HH, nullptr, Y, H, N,
        perm, tType, tStart, tCnt, numTiles, 1);
    float* sw = h; h = hn; hn = sw;
  }
}